// CoreRNNFW_65463891525848
// MI455X (gfx1250) — compile-verified
//
#include <hip/hip_runtime.h>
#include <hip/hip_bf16.h>
#include <math.h>

// ---------------- problem dims (match reference) ----------------
#define T_STEPS 24
#define T_MID   23          // mid steps 0..22; step 23 is the retrieval step
#define BB      16
#define DG      256
#define DH      768
#define DOUT    256
#define SINNER  4
#define NWG     16
#define NTHR    256
#define NCOLW   48          // output columns owned per WG (768/16)
#define KSPLIT  512         // K split point between wave j and wave j+3

#define LAM_C   0.95f
#define ETA_C   0.5f
#define EPS_C   1e-6f
#define LNEPS_C 1e-5f

// padded LDS strides (dwords) so 16 lanes at column-stride hit distinct banks
#define WH_STRIDE 770       // >= 768, even (b64 aligned), 770 % 64 == 2
#define WG_STRIDE 258       // >= 256, even,               258 % 64 == 2

typedef __attribute__((ext_vector_type(2))) float v2f;
typedef __attribute__((ext_vector_type(8))) float v8f;

// ---------------- workspace layout (float units) ----------------
#define WS_HCUR   0                         // [16 x 768]  current h (row-major [b][k])
#define WS_HBASE  (WS_HCUR  + BB*DH)        // [16 x 768]  h_base of current step
#define WS_HIST   (WS_HBASE + BB*DH)        // [16][23][768] history of h (per sample)
#define WS_REL    (WS_HIST  + BB*T_MID*DH)  // [16] per-sample log1p(rel_err)
#define WS_SYNC   (WS_REL   + 16)           // 2 x u32: barrier counter / generation

// ---------------- grid barrier over 16 resident WGs ----------------
__device__ __forceinline__ void grid_barrier(unsigned* cnt, unsigned* gen) {
    __syncthreads();
    if (threadIdx.x == 0) {
        __threadfence();
        unsigned g = atomicAdd(gen, 0u);
        if (atomicAdd(cnt, 1u) == NWG - 1u) {
            atomicExch(cnt, 0u);
            atomicAdd(gen, 1u);
        } else {
            while (atomicAdd(gen, 0u) == g) { __builtin_amdgcn_s_sleep(2); }
        }
        __threadfence();
    }
    __syncthreads();
}

// ---------------- reductions (wave32) ----------------
__device__ __forceinline__ float waveRed(float v) {
    #pragma unroll
    for (int m = 16; m > 0; m >>= 1) v += __shfl_xor(v, m, 32);
    return v;
}
__device__ __forceinline__ float blockRed(float v, volatile float* s8) {
    const int w = threadIdx.x >> 5, l = threadIdx.x & 31;
    v = waveRed(v);
    __syncthreads();
    if (l == 0) s8[w] = v;
    __syncthreads();
    float r = s8[0] + s8[1] + s8[2] + s8[3] + s8[4] + s8[5] + s8[6] + s8[7];
    __syncthreads();
    return r;
}

__global__ void __launch_bounds__(NTHR)
fastweight_init_kernel(unsigned* sync) {
    sync[0] = 0u;
    sync[1] = 0u;
}

__global__ void __launch_bounds__(NTHR)
fastweight_persistent_kernel(const float* __restrict__ z_seq,   // [24,16,256]
                             const float* __restrict__ clean,   // [16,256]
                             const float* __restrict__ W_h,     // [768,768]
                             const float* __restrict__ W_g,     // [768,256]
                             const float* __restrict__ b_h,     // [768]
                             const float* __restrict__ alpha_p, // [1]
                             const float* __restrict__ ln_g,    // [768]
                             const float* __restrict__ ln_b,    // [768]
                             const float* __restrict__ head_W,  // [256,768]
                             const float* __restrict__ head_b,  // [256]
                             float* __restrict__ out,           // [1]
                             float* __restrict__ ws) {
    const int tid  = threadIdx.x;
    const int wg   = blockIdx.x;           // 0..15 (= batch sample for inner phase)
    const int wave = tid >> 5;
    const int lane = tid & 31;
    const int lo   = lane & 15;
    const int hi   = lane >> 4;

    float*    hcur   = ws + WS_HCUR;
    float*    hbaseg = ws + WS_HBASE;
    float*    histb  = ws + WS_HIST + (size_t)wg * T_MID * DH;
    float*    relv   = ws + WS_REL;
    unsigned* sync   = (unsigned*)(ws + WS_SYNC);

    // ---- big LDS: per-WG weight slices, resident for the entire kernel ----
    __shared__ float sWh[NCOLW * WH_STRIDE];   // 147.8 KB  (48 cols x 768 K, padded)
    __shared__ float sWg[NCOLW * WG_STRIDE];   //  49.5 KB  (48 cols x 256 K, padded)
    __shared__ float sPart[3 * 32 * 8];        // partial C tiles from K-split waves
    __shared__ float sh_hbase[DH];
    __shared__ float sh_hs[DH];
    __shared__ float sh_Ah[DH];
    __shared__ float sh_tmp[DH];
    __shared__ float sh_d[T_MID + 1];
    __shared__ float sh_coef[T_MID + 1];
    __shared__ float sRed[8];

    // k = where(alpha>=0, 1+softplus(alpha), 1/(1+softplus(-alpha)))
    const float alpha = alpha_p[0];
    const float kval  = (alpha >= 0.0f) ? (1.0f + log1pf(expf(alpha)))
                                        : (1.0f / (1.0f + log1pf(expf(-alpha))));

    // ---- one-time fill: this WG's 48-column slices of W_h / W_g -> LDS ----
    {
        const float2* Wh2 = (const float2*)(W_h + (size_t)wg * NCOLW * DH);
        for (int idx = tid; idx < NCOLW * (DH / 2); idx += NTHR) {
            const int c = idx / (DH / 2), k2 = idx % (DH / 2);
            *(float2*)(sWh + c * WH_STRIDE + 2 * k2) = Wh2[(size_t)c * (DH / 2) + k2];
        }
        const float2* Wg2 = (const float2*)(W_g + (size_t)wg * NCOLW * DG);
        for (int idx = tid; idx < NCOLW * (DG / 2); idx += NTHR) {
            const int c = idx / (DG / 2), k2 = idx % (DG / 2);
            *(float2*)(sWg + c * WG_STRIDE + 2 * k2) = Wg2[(size_t)c * (DG / 2) + k2];
        }
    }

    // init: zero h0 slice (disjoint per WG) + decay coefficients
    for (int i = tid; i < DH; i += NTHR) hcur[(size_t)wg * DH + i] = 0.0f;
    if (tid <= T_MID) sh_coef[tid] = 0.0f;
    grid_barrier(sync, sync + 1);   // also covers the LDS fill (__syncthreads inside)

    for (int t = 0; t < T_STEPS; ++t) {
        // ======== phase 1: h_base = h @ W_h^T + z_t @ W_g^T + b_h ========
        // WG w owns output columns [48w, 48w+48); 6 waves: tile j handled by
        // wave j (W_h K in [0,512)) + wave j+3 (W_h K in [512,768) + all W_g).
        // Separate branches with constant trip counts so every K-loop gets
        // unrolled 16x + software-pipelined (loads overlap WMMA).
        v8f acc = {};
        const int j = (wave < 6) ? (wave % 3) : 0;
        if (wave < 3) {
            const float* Ah_row = hcur + (size_t)lo * DH;                // A: rows of h
            const float* Bh_col = sWh + (j * 16 + lo) * WH_STRIDE;       // B[k][n] = W_h[n][k] (LDS)
            #pragma unroll 16
            for (int k0 = 0; k0 < KSPLIT; k0 += 4) {
                v2f a, b;
                a.x = Ah_row[k0 + 2 * hi];     a.y = Ah_row[k0 + 2 * hi + 1];
                b.x = Bh_col[k0 + 2 * hi];     b.y = Bh_col[k0 + 2 * hi + 1];
                acc = __builtin_amdgcn_wmma_f32_16x16x4_f32(
                        false, a, false, b, (short)0, acc, false, false);
            }
        } else if (wave < 6) {
            const float* Ah_row = hcur + (size_t)lo * DH;
            const float* Bh_col = sWh + (j * 16 + lo) * WH_STRIDE;
            #pragma unroll 16
            for (int k0 = KSPLIT; k0 < DH; k0 += 4) {
                v2f a, b;
                a.x = Ah_row[k0 + 2 * hi];     a.y = Ah_row[k0 + 2 * hi + 1];
                b.x = Bh_col[k0 + 2 * hi];     b.y = Bh_col[k0 + 2 * hi + 1];
                acc = __builtin_amdgcn_wmma_f32_16x16x4_f32(
                        false, a, false, b, (short)0, acc, false, false);
            }
            const float* Az_row = z_seq + (size_t)t * BB * DG + (size_t)lo * DG;
            const float* Bg_col = sWg + (j * 16 + lo) * WG_STRIDE;
            #pragma unroll 16
            for (int k0 = 0; k0 < DG; k0 += 4) {
                v2f a, b;
                a.x = Az_row[k0 + 2 * hi];     a.y = Az_row[k0 + 2 * hi + 1];
                b.x = Bg_col[k0 + 2 * hi];     b.y = Bg_col[k0 + 2 * hi + 1];
                acc = __builtin_amdgcn_wmma_f32_16x16x4_f32(
                        false, a, false, b, (short)0, acc, false, false);
            }
        }
        __syncthreads();
        if (wave >= 3 && wave < 6) {          // publish the high-K partials
            #pragma unroll
            for (int r = 0; r < 8; ++r) sPart[(j * 32 + lane) * 8 + r] = acc[r];
        }
        __syncthreads();
        if (wave < 3) {                       // combine + bias + writeback
            const int col = wg * NCOLW + j * 16 + lo;
            const float bc = b_h[col];
            #pragma unroll
            for (int r = 0; r < 8; ++r) {
                const int row = r + 8 * hi;   // C layout: VGPR r -> M=r (lanes 0-15), M=r+8 (16-31)
                hbaseg[(size_t)row * DH + col] = acc[r] + sPart[(j * 32 + lane) * 8 + r] + bc;
            }
        }
        // prefetch next timestep's z slab while we still have memory slack
        if (t + 1 < T_STEPS) {
            __builtin_prefetch(z_seq + (size_t)(t + 1) * BB * DG + tid * 16, 0, 1);
        }
        grid_barrier(sync, sync + 1);

        // ======== phase 2: inner loop — WG b owns sample b ========
        for (int i = tid; i < DH; i += NTHR) {
            const float v = hbaseg[(size_t)wg * DH + i];
            sh_hbase[i] = v;
            sh_hs[i]    = fmaxf(v, 0.0f);
        }
        __syncthreads();

        const int nh = t;  // history entries available (A after t updates)
        for (int it = 0; it < SINNER; ++it) {
            // d_tau = coef_tau * <hist_tau, h_s>   (waves own taus round-robin)
            for (int tau = wave; tau < nh; tau += 8) {
                const float* hr = histb + (size_t)tau * DH;
                float p = 0.0f;
                for (int i = lane; i < DH; i += 32) p += hr[i] * sh_hs[i];
                p = waveRed(p);
                if (lane == 0) sh_d[tau] = p * sh_coef[tau];
            }
            __syncthreads();
            // Ah_i = sum_tau d_tau * hist_tau[i]
            for (int i = tid; i < DH; i += NTHR) {
                float ah = 0.0f;
                for (int tau = 0; tau < nh; ++tau) ah += sh_d[tau] * histb[(size_t)tau * DH + i];
                sh_Ah[i] = ah;
            }
            __syncthreads();

            if (t < T_MID) {
                float pd = 0.0f, p1 = 0.0f, p2 = 0.0f;
                for (int i = tid; i < DH; i += NTHR) {
                    const float hs = sh_hs[i], ah = sh_Ah[i];
                    pd += hs * ah; p1 += hs * hs; p2 += ah * ah;
                }
                const float dot = blockRed(pd, sRed);
                const float n1s = blockRed(p1, sRed);
                const float n2s = blockRed(p2, sRed);
                const float n1 = sqrtf(n1s) + 1e-6f;
                const float n2 = sqrtf(n2s) + 1e-6f;
                float R = dot / (n1 * n2 + 1e-6f);
                R = fminf(fmaxf(R, 0.0f), 1.0f);
                const float a    = 1.0f - powf(1.0f - R, kval);
                const float oma2 = 1.0f - a * a;
                for (int i = tid; i < DH; i += NTHR)
                    sh_tmp[i] = oma2 * sh_hbase[i] + a * sh_Ah[i];
            } else {
                for (int i = tid; i < DH; i += NTHR)
                    sh_tmp[i] = sh_hbase[i] + sh_Ah[i];
            }
            __syncthreads();

            // LayerNorm + relu
            float ps = 0.0f, pq = 0.0f;
            for (int i = tid; i < DH; i += NTHR) {
                const float v = sh_tmp[i];
                ps += v; pq += v * v;
            }
            const float sum   = blockRed(ps, sRed);
            const float sumsq = blockRed(pq, sRed);
            const float mu   = sum * (1.0f / DH);
            const float var  = sumsq * (1.0f / DH) - mu * mu;
            const float rstd = rsqrtf(var + LNEPS_C);
            for (int i = tid; i < DH; i += NTHR) {
                const float v = ln_g[i] * (sh_tmp[i] - mu) * rstd + ln_b[i];
                sh_hs[i] = fmaxf(v, 0.0f);
            }
            __syncthreads();
        }

        if (t < T_MID) {
            // Hebbian update folded into the low-rank history:
            // coef_tau *= lambda (tau<t); coef_t = eta / (||h||^2 + eps)
            float pn = 0.0f;
            for (int i = tid; i < DH; i += NTHR) pn += sh_hs[i] * sh_hs[i];
            const float hn2 = blockRed(pn, sRed) + EPS_C;
            for (int i = tid; i < DH; i += NTHR) {
                const float v = sh_hs[i];
                histb[(size_t)t * DH + i]   = v;
                hcur[(size_t)wg * DH + i]   = v;
            }
            if (tid == 0) {
                for (int tau = 0; tau < t; ++tau) sh_coef[tau] *= LAM_C;
                sh_coef[t] = ETA_C / hn2;
            }
            __syncthreads();
        }
        grid_barrier(sync, sync + 1);
    }

    // ======== head + loss: WG b computes log1p(rel_err_b) ========
    {
        const int o = tid;                       // NTHR == DOUT == 256
        const float* Wr = head_W + (size_t)o * DH;
        float pred = head_b[o];
        for (int i = 0; i < DH; ++i) pred += Wr[i] * sh_hs[i];
        const float cv = clean[(size_t)wg * DOUT + o];
        const float df = pred - cv;
        const float se = blockRed(df * df, sRed);
        const float nc = blockRed(cv * cv, sRed);
        if (tid == 0) relv[wg] = log1pf(se / (nc + 1e-6f));
    }
    grid_barrier(sync, sync + 1);

    if (wg == 0 && tid == 0) {
        float acc = 0.0f;
        for (int b = 0; b < BB; ++b) acc += relv[b];   // deterministic in-order sum
        out[0] = acc * (1.0f / BB);
    }
}

extern "C" void kernel_launch(void* const* d_in, const int* in_sizes, int n_in,
                              void* d_out, int out_size, void* d_ws, size_t ws_size,
                              hipStream_t stream) {
    const float* z_seq  = (const float*)d_in[0];
    const float* clean  = (const float*)d_in[1];
    const float* W_h    = (const float*)d_in[2];
    const float* W_g    = (const float*)d_in[3];
    const float* b_h    = (const float*)d_in[4];
    const float* alpha  = (const float*)d_in[5];
    const float* ln_g   = (const float*)d_in[6];
    const float* ln_b   = (const float*)d_in[7];
    const float* head_W = (const float*)d_in[8];
    const float* head_b = (const float*)d_in[9];
    float*       out    = (float*)d_out;
    float*       ws     = (float*)d_ws;

    fastweight_init_kernel<<<1, NTHR, 0, stream>>>((unsigned*)(ws + WS_SYNC));
    fastweight_persistent_kernel<<<NWG, NTHR, 0, stream>>>(
        z_seq, clean, W_h, W_g, b_h, alpha, ln_g, ln_b, head_W, head_b, out, ws);
}